// SimpleRNN_23605140259184
// MI455X (gfx1250) — compile-verified
//
#include <hip/hip_runtime.h>
#include <stdint.h>

#define RNN_H   8
#define RNN_T   1024
#define BLK     128
#define CH      32              // timesteps per prefetched chunk (128 B / lane)
#define NCH     (RNN_T / CH)    // 32 chunks
#define ROW     (CH + 4)        // padded LDS row: 36 floats = 144 B (16B-aligned stride)

// ---- gfx1250 hardware tanh (transcendental VALU op) ----
__device__ __forceinline__ float hw_tanhf(float x) {
#if defined(__has_builtin) && __has_builtin(__builtin_amdgcn_tanhf)
  return __builtin_amdgcn_tanhf(x);
#else
  float r;
  asm volatile("v_tanh_f32 %0, %1" : "=v"(r) : "v"(x));
  return r;
#endif
}

__device__ __forceinline__ void wait_async0() {
#if defined(__has_builtin) && __has_builtin(__builtin_amdgcn_s_wait_asynccnt)
  __builtin_amdgcn_s_wait_asynccnt(0);
#else
  asm volatile("s_wait_asynccnt 0x0" ::: "memory");
#endif
}

// Async DMA of one 128-byte per-lane row (32 floats) from global to LDS.
// vdst = VGPR holding LDS byte address; vaddr = 64-bit per-lane global address.
// The same INST_OFFSET advances both the LDS and global addresses (ISA 15.18.3).
__device__ __forceinline__ void async_copy_row128(uint32_t lds, const float* g) {
  uint64_t ga = (uint64_t)(uintptr_t)g;
  asm volatile(
      "global_load_async_to_lds_b128 %0, %1, off\n\t"
      "global_load_async_to_lds_b128 %0, %1, off offset:16\n\t"
      "global_load_async_to_lds_b128 %0, %1, off offset:32\n\t"
      "global_load_async_to_lds_b128 %0, %1, off offset:48\n\t"
      "global_load_async_to_lds_b128 %0, %1, off offset:64\n\t"
      "global_load_async_to_lds_b128 %0, %1, off offset:80\n\t"
      "global_load_async_to_lds_b128 %0, %1, off offset:96\n\t"
      "global_load_async_to_lds_b128 %0, %1, off offset:112"
      :: "v"(lds), "v"(ga) : "memory");
}

__global__ __launch_bounds__(BLK, 1)
void rnn_fused_kernel(const float* __restrict__ input,   // [B, T, 1]
                      const float* __restrict__ W_ih,    // [H, 1]
                      const float* __restrict__ W_hh,    // [H, H]
                      const float* __restrict__ b_ih,    // [H]
                      const float* __restrict__ b_hh,    // [H]
                      const float* __restrict__ fc_w,    // [1, H]
                      const float* __restrict__ fc_b,    // [1]
                      float* __restrict__ out)           // [B]
{
  __shared__ float sbuf[2][BLK][ROW];

  const int tid = threadIdx.x;
  const int b   = blockIdx.x * BLK + tid;   // one lane = one batch element

  // ---- uniform weights: scalar loads -> SGPR operands ----
  float w[RNN_H][RNN_H];
  float wih[RNN_H], bsum[RNN_H];
#pragma unroll
  for (int j = 0; j < RNN_H; ++j) {
    wih[j]  = W_ih[j];
    bsum[j] = b_ih[j] + b_hh[j];
#pragma unroll
    for (int k = 0; k < RNN_H; ++k) w[j][k] = W_hh[j * RNN_H + k];
  }

  const float* gsrc = input + (size_t)b * RNN_T;
  const uint32_t lds0 = (uint32_t)(uintptr_t)(&sbuf[0][tid][0]);
  const uint32_t lds1 = (uint32_t)(uintptr_t)(&sbuf[1][tid][0]);

  // prologue: async-prefetch chunk 0 into buffer 0
  async_copy_row128(lds0, gsrc);

  float h[RNN_H];
#pragma unroll
  for (int j = 0; j < RNN_H; ++j) h[j] = 0.0f;

  for (int c = 0; c < NCH; ++c) {
    const int cur = c & 1;

    // wait for chunk c's async LDS writes to land
    wait_async0();

    // kick off async prefetch of chunk c+1 into the other buffer
    // (its previous reader finished an iteration ago -> no WAR hazard)
    if (c + 1 < NCH) {
      async_copy_row128(cur ? lds0 : lds1, gsrc + (size_t)(c + 1) * CH);
    }

    // pull this lane's 32 inputs from LDS into registers (ds_load_b128 x8)
    float x[CH];
#pragma unroll
    for (int wd = 0; wd < CH / 4; ++wd) {
      const float4 v = *(const float4*)&sbuf[cur][tid][wd * 4];
      x[wd * 4 + 0] = v.x;
      x[wd * 4 + 1] = v.y;
      x[wd * 4 + 2] = v.z;
      x[wd * 4 + 3] = v.w;
    }

    // 32 recurrence steps; 8 independent FMA chains (ILP=8), hw tanh
#pragma unroll
    for (int t = 0; t < CH; ++t) {
      float a[RNN_H];
#pragma unroll
      for (int j = 0; j < RNN_H; ++j) a[j] = fmaf(x[t], wih[j], bsum[j]);
#pragma unroll
      for (int k = 0; k < RNN_H; ++k) {
#pragma unroll
        for (int j = 0; j < RNN_H; ++j) a[j] = fmaf(w[j][k], h[k], a[j]);
      }
#pragma unroll
      for (int j = 0; j < RNN_H; ++j) h[j] = hw_tanhf(a[j]);
    }
  }

  // linear head: y = h . fc_w + fc_b
  float y = fc_b[0];
#pragma unroll
  for (int j = 0; j < RNN_H; ++j) y = fmaf(h[j], fc_w[j], y);
  out[b] = y;
}

extern "C" void kernel_launch(void* const* d_in, const int* in_sizes, int n_in,
                              void* d_out, int out_size, void* d_ws, size_t ws_size,
                              hipStream_t stream) {
  (void)n_in; (void)d_ws; (void)ws_size; (void)out_size;
  const float* input = (const float*)d_in[0];
  const float* W_ih  = (const float*)d_in[1];
  const float* W_hh  = (const float*)d_in[2];
  const float* b_ih  = (const float*)d_in[3];
  const float* b_hh  = (const float*)d_in[4];
  const float* fc_w  = (const float*)d_in[5];
  const float* fc_b  = (const float*)d_in[6];
  float* out = (float*)d_out;

  const int B = in_sizes[0] / RNN_T;   // 16384
  dim3 grid(B / BLK), block(BLK);
  rnn_fused_kernel<<<grid, block, 0, stream>>>(input, W_ih, W_hh, b_ih, b_hh,
                                               fc_w, fc_b, out);
}